// WeightModulatedINR_51410758533754
// MI455X (gfx1250) — compile-verified
//
#include <hip/hip_runtime.h>
#include <hip/hip_bf16.h>

typedef __attribute__((ext_vector_type(2))) float v2f;
typedef __attribute__((ext_vector_type(8))) float v8f;

__device__ __forceinline__ v8f wmma4(v2f a, v2f b, v8f c) {
  // D = A(16x4 f32) * B(4x16 f32) + C(16x16 f32)
  return __builtin_amdgcn_wmma_f32_16x16x4_f32(false, a, false, b, (short)0, c, false, false);
}

#define FEAT_SPATIAL 7680          // 16*16*30 voxels per batch
#define FEAT_BSTRIDE (256 * 7680)  // per-batch stride of feat0

// ---------------------------------------------------------------------------
// Kernel 1: fused conv1 (256->128, relu) + conv2 (128->64, relu) per 16-row
// tile. f1 lives only in LDS; f2 -> global workspace [61440, 64].
// A tile stored K-pair-interleaved so fragment loads are single ds_load_b64.
// ---------------------------------------------------------------------------
__global__ __launch_bounds__(256) void conv12_kernel(
    const float* __restrict__ feat0,
    const float* __restrict__ cw1, const float* __restrict__ cb1,
    const float* __restrict__ cw2, const float* __restrict__ cb2,
    float* __restrict__ f2) {
  __shared__ float A_lds[256 * 16];     // interleaved: [(c>>1)*32 + r*2 + (c&1)]
  __shared__ float f1_lds[16 * 132];    // [m][k], stride 132 (even, bank-spread 4)

  const int tid  = threadIdx.x;
  const int lane = tid & 31;
  const int wave = tid >> 5;
  const int half = lane >> 4;
  const int l16  = lane & 15;

  const int m0 = blockIdx.x * 16;               // tiles never cross batch (7680 % 16 == 0)
  const int b  = m0 / FEAT_SPATIAL;
  const int s0 = m0 % FEAT_SPATIAL;
  const float* Abase = feat0 + (size_t)b * FEAT_BSTRIDE + s0;

  // cooperative, coalesced load of A tile (K-pair interleaved in LDS)
  #pragma unroll
  for (int it = 0; it < 16; ++it) {
    int i = it * 256 + tid;
    int c = i >> 4, r = i & 15;
    A_lds[(c >> 1) * 32 + r * 2 + (c & 1)] = Abase[(size_t)c * FEAT_SPATIAL + r];
  }
  __syncthreads();

  // conv1: wave w computes output cols [w*16, w*16+16), K = 256
  {
    const int n0 = wave * 16;
    const float bias = cb1[n0 + l16];
    v8f acc;
    #pragma unroll
    for (int r = 0; r < 8; ++r) acc[r] = bias;
    #pragma unroll 4
    for (int k0 = 0; k0 < 256; k0 += 4) {
      const int ca = k0 + half * 2;  // even
      v2f a = *(const v2f*)&A_lds[(ca >> 1) * 32 + l16 * 2];           // ds_load_b64
      v2f bm; bm.x = cw1[ca * 128 + n0 + l16];   bm.y = cw1[(ca + 1) * 128 + n0 + l16];
      acc = wmma4(a, bm, acc);
    }
    #pragma unroll
    for (int r = 0; r < 8; ++r) {
      float v = acc[r] > 0.f ? acc[r] : 0.f;     // ReLU
      f1_lds[(r + half * 8) * 132 + n0 + l16] = v;
    }
  }
  __syncthreads();

  // conv2: waves 0..3 compute output cols [w*16, w*16+16), K = 128
  if (wave < 4) {
    const int n0 = wave * 16;
    const float bias = cb2[n0 + l16];
    v8f acc;
    #pragma unroll
    for (int r = 0; r < 8; ++r) acc[r] = bias;
    #pragma unroll 4
    for (int k0 = 0; k0 < 128; k0 += 4) {
      const int ca = k0 + half * 2;  // even -> 8B aligned
      v2f a = *(const v2f*)&f1_lds[l16 * 132 + ca];                    // ds_load_b64
      v2f bm; bm.x = cw2[ca * 64 + n0 + l16];    bm.y = cw2[(ca + 1) * 64 + n0 + l16];
      acc = wmma4(a, bm, acc);
    }
    #pragma unroll
    for (int r = 0; r < 8; ++r) {
      float v = acc[r] > 0.f ? acc[r] : 0.f;     // ReLU
      f2[(size_t)(m0 + r + half * 8) * 64 + n0 + l16] = v;
    }
  }
}

// ---------------------------------------------------------------------------
// Kernel 2: conv3 on stride-2 subset (64->32, relu). 7680 selected voxels.
// Output stored transposed + K-pair interleaved for the hypernet GEMM:
// At2[(j>>1)*32 + b*2 + (j&1)], j = c*960 + t  (torch [B,C,D,H,W] flatten).
// ---------------------------------------------------------------------------
__global__ __launch_bounds__(64) void conv3_kernel(
    const float* __restrict__ f2,
    const float* __restrict__ cw3, const float* __restrict__ cb3,
    float* __restrict__ At2) {
  const int tid  = threadIdx.x;
  const int lane = tid & 31;
  const int wave = tid >> 5;
  const int half = lane >> 4;
  const int l16  = lane & 15;

  const int mt = blockIdx.x;         // 0..479, 960 selected rows per batch / 16
  const int b  = mt / 60;
  const int t0 = (mt % 60) * 16;
  const int n0 = wave * 16;          // N = 32 -> 2 waves

  // this lane's A-row: selected index t -> original voxel v (d,h,w even)
  const int t  = t0 + l16;
  const int d2 = t / 120, rem = t % 120, h2 = rem / 15, w2 = rem % 15;
  const size_t v = (size_t)b * 7680 + d2 * 960 + h2 * 60 + 2 * w2;
  const float* arow = f2 + v * 64;

  const float bias = cb3[n0 + l16];
  v8f acc;
  #pragma unroll
  for (int r = 0; r < 8; ++r) acc[r] = bias;
  #pragma unroll
  for (int k0 = 0; k0 < 64; k0 += 4) {
    const int ca = k0 + half * 2;
    v2f a = *(const v2f*)(arow + ca);            // contiguous pair, 8B aligned
    v2f bm; bm.x = cw3[ca * 32 + n0 + l16];      bm.y = cw3[(ca + 1) * 32 + n0 + l16];
    acc = wmma4(a, bm, acc);
  }
  const int c = n0 + l16;
  #pragma unroll
  for (int r = 0; r < 8; ++r) {
    float val = acc[r] > 0.f ? acc[r] : 0.f;     // ReLU
    const int tsel = t0 + r + half * 8;
    const size_t j = (size_t)c * 960 + tsel;
    At2[(j >> 1) * 32 + b * 2 + (j & 1)] = val;
  }
}

// ---------------------------------------------------------------------------
// Kernel 3: hypernet GEMM wmod = A(8x30720, padded to 16) @ hw(30720x7680).
// Split-K (16 ways) with deterministic workspace partials. Each wave handles
// TWO adjacent 16-col tiles: 128 consecutive bytes per hw row (full line),
// one A-fragment feeding two WMMAs. hw streamed once (944 MB) nontemporally.
// ---------------------------------------------------------------------------
__global__ __launch_bounds__(256) void hyper_kernel(
    const float* __restrict__ At2,
    const float* __restrict__ hw,
    float* __restrict__ part) {
  const int tid  = threadIdx.x;
  const int lane = tid & 31;
  const int wave = tid >> 5;
  const int half = lane >> 4;
  const int l16  = lane & 15;

  const int n0 = blockIdx.x * 256 + wave * 32;   // 240 wave-tiles of 32 cols
  const int ks = blockIdx.y * 1920;              // 16 K-splits of 1920

  v8f acc0, acc1;
  #pragma unroll
  for (int r = 0; r < 8; ++r) { acc0[r] = 0.f; acc1[r] = 0.f; }

  const float* hcol = hw + n0 + l16;
  #pragma unroll 4
  for (int k0 = ks; k0 < ks + 1920; k0 += 4) {
    const int ca = k0 + half * 2;  // even
    v2f a = *(const v2f*)&At2[(size_t)(ca >> 1) * 32 + l16 * 2];       // b64, L2-hot
    const float* hr0 = hcol + (size_t)ca * 7680;
    const float* hr1 = hr0 + 7680;
    v2f b0, b1;
    b0.x = __builtin_nontemporal_load(hr0);
    b0.y = __builtin_nontemporal_load(hr1);
    b1.x = __builtin_nontemporal_load(hr0 + 16);
    b1.y = __builtin_nontemporal_load(hr1 + 16);
    acc0 = wmma4(a, b0, acc0);
    acc1 = wmma4(a, b1, acc1);
  }

  // only D rows 0..7 are real (lanes 0-15 hold M = r)
  if (lane < 16) {
    #pragma unroll
    for (int r = 0; r < 8; ++r) {
      part[((size_t)blockIdx.y * 8 + r) * 7680 + n0 + l16]      = acc0[r];
      part[((size_t)blockIdx.y * 8 + r) * 7680 + n0 + 16 + l16] = acc1[r];
    }
  }
}

// Kernel 3b: deterministic split-K reduction + bias
__global__ __launch_bounds__(256) void reduce_kernel(
    const float* __restrict__ part, const float* __restrict__ hb,
    float* __restrict__ wmod) {
  const int i = blockIdx.x * 256 + threadIdx.x;  // 0..61439
  if (i < 8 * 7680) {
    const int b = i / 7680, n = i % 7680;
    float s = hb[n];
    #pragma unroll
    for (int sp = 0; sp < 16; ++sp) s += part[(size_t)(sp * 8 + b) * 7680 + n];
    wmod[i] = s;
  }
}

// ---------------------------------------------------------------------------
// Kernel 4: Fourier features + modulated bmm + 2x(Linear+ReLU) + head.
// One block per batch element; trivially small.
// ---------------------------------------------------------------------------
__global__ __launch_bounds__(128) void mlp_kernel(
    const float* __restrict__ coord, const float* __restrict__ wmod,
    const float* __restrict__ w1, const float* __restrict__ b1,
    const float* __restrict__ w2, const float* __restrict__ b2,
    const float* __restrict__ wo, const float* __restrict__ bo,
    float* __restrict__ out) {
  __shared__ float zf[120];
  __shared__ float ha[64];
  __shared__ float hb_s[64];
  const int b = blockIdx.x;
  const int t = threadIdx.x;

  if (t < 120) {
    const int i = t / 40, j = t % 40;
    // freqs = exp(linspace(0, ln(1024), 40))
    const float freq = __expf((float)j * (6.93147180559945f / 39.0f));
    zf[t] = coord[b * 3 + i] * freq;
  }
  __syncthreads();

  if (t < 64) {  // modulated layer 0: no bias, no activation
    float acc = 0.f;
    #pragma unroll 4
    for (int f = 0; f < 120; ++f) acc += zf[f] * wmod[(size_t)b * 7680 + f * 64 + t];
    ha[t] = acc;
  }
  __syncthreads();

  if (t < 64) {  // layer 1 + ReLU
    float acc = b1[t];
    #pragma unroll 4
    for (int j = 0; j < 64; ++j) acc += ha[j] * w1[j * 64 + t];
    hb_s[t] = acc > 0.f ? acc : 0.f;
  }
  __syncthreads();

  if (t < 64) {  // layer 2 + ReLU
    float acc = b2[t];
    #pragma unroll 4
    for (int j = 0; j < 64; ++j) acc += hb_s[j] * w2[j * 64 + t];
    ha[t] = acc > 0.f ? acc : 0.f;
  }
  __syncthreads();

  if (t < 3) {   // output head
    float acc = bo[t];
    #pragma unroll 4
    for (int j = 0; j < 64; ++j) acc += ha[j] * wo[j * 3 + t];
    out[b * 3 + t] = acc;
  }
}

// ---------------------------------------------------------------------------
extern "C" void kernel_launch(void* const* d_in, const int* in_sizes, int n_in,
                              void* d_out, int out_size, void* d_ws, size_t ws_size,
                              hipStream_t stream) {
  const float* coord = (const float*)d_in[0];
  const float* feat0 = (const float*)d_in[1];
  const float* cw1   = (const float*)d_in[2];
  const float* cb1   = (const float*)d_in[3];
  const float* cw2   = (const float*)d_in[4];
  const float* cb2   = (const float*)d_in[5];
  const float* cw3   = (const float*)d_in[6];
  const float* cb3   = (const float*)d_in[7];
  const float* hw    = (const float*)d_in[8];
  const float* hb    = (const float*)d_in[9];
  const float* w1    = (const float*)d_in[10];
  const float* b1    = (const float*)d_in[11];
  const float* w2    = (const float*)d_in[12];
  const float* b2    = (const float*)d_in[13];
  const float* wo    = (const float*)d_in[14];
  const float* bo    = (const float*)d_in[15];
  float* out = (float*)d_out;

  // workspace layout (floats): f2[61440*64] | At2[30720*16] | part[16*8*7680] | wmod[8*7680]
  float* f2   = (float*)d_ws;
  float* At2  = f2 + (size_t)61440 * 64;
  float* part = At2 + (size_t)30720 * 16;
  float* wmod = part + (size_t)16 * 8 * 7680;

  conv12_kernel<<<3840, 256, 0, stream>>>(feat0, cw1, cb1, cw2, cb2, f2);
  conv3_kernel<<<480, 64, 0, stream>>>(f2, cw3, cb3, At2);
  hyper_kernel<<<dim3(30, 16), 256, 0, stream>>>(At2, hw, part);
  reduce_kernel<<<240, 256, 0, stream>>>(part, hb, wmod);
  mlp_kernel<<<8, 128, 0, stream>>>(coord, wmod, w1, b1, w2, b2, wo, bo, out);
}